// IPGNN_31593779429379
// MI455X (gfx1250) — compile-verified
//
#include <hip/hip_runtime.h>
#include <hip/hip_bf16.h>
#include <math.h>

// Problem constants (from reference)
#define NN   6000
#define EE   192000
#define INF  19900
#define HH   16
#define HE   128
#define LL   4
#define CC   2
#define BN_EPS 1e-5f
#define COS_EPS 1e-8f

// Workspace float offsets
#define OFF_MU      0      // mu[4]
#define OFF_RSD     4      // 1/sd[4]
#define OFF_ESUM    8      // edge-BN sum [2][128]
#define OFF_ESQ     264    // edge-BN sumsq [2][128]
#define OFF_ESCALE  520    // folded BN scale [2][128]
#define OFF_ESHIFT  776    // folded BN shift [2][128]
#define OFF_WSM     1032   // softmax(layer_w)[4]
#define OFF_BNSCALE 1040   // per-layer GCN BN scale[16]
#define OFF_BNSHIFT 1056   // per-layer GCN BN shift[16]
#define OFF_BIG     2048

typedef float v2f __attribute__((ext_vector_type(2)));
typedef float v8f __attribute__((ext_vector_type(8)));

#if defined(__has_builtin)
#if __has_builtin(__builtin_amdgcn_global_load_async_to_lds_b128) && \
    __has_builtin(__builtin_amdgcn_s_wait_asynccnt)
#define HAVE_ASYNC 1
#endif
#endif

typedef int v4i __attribute__((vector_size(4 * sizeof(int))));
typedef __attribute__((address_space(1))) v4i as1_v4i;
typedef __attribute__((address_space(3))) v4i as3_v4i;

// 16-byte global -> LDS copy; async (ASYNCcnt-tracked) on gfx1250.
__device__ __forceinline__ void cp16_async(const float* g, float* l) {
#ifdef HAVE_ASYNC
  __builtin_amdgcn_global_load_async_to_lds_b128((as1_v4i*)g, (as3_v4i*)l, 0,
                                                 0);
#else
  *(float4*)l = *(const float4*)g;
#endif
}

__device__ __forceinline__ void async_wait0() {
#ifdef HAVE_ASYNC
  __builtin_amdgcn_s_wait_asynccnt(0);
#endif
}

__device__ __forceinline__ v8f wmma4(v2f a, v2f b, v8f c) {
  // V_WMMA_F32_16X16X4_F32 : D(16x16,f32) = A(16x4,f32) x B(4x16,f32) + C
  return __builtin_amdgcn_wmma_f32_16x16x4_f32(false, a, false, b, (short)0, c,
                                               false, false);
}

// ---------------------------------------------------------------- utilities
__global__ void k_zero(float* p, int n) {
  int t = blockIdx.x * blockDim.x + threadIdx.x;
  if (t < n) p[t] = 0.f;
}

// ------------------------------------------------ 1) edge_input column stats
// mean + unbiased std (ddof=1) over E rows, 4 columns; single block.
__global__ void k_edge_std(const float* __restrict__ einp, float* __restrict__ W) {
  __shared__ float red[8][256];
  int t = threadIdx.x;
  float s[4] = {0, 0, 0, 0}, q[4] = {0, 0, 0, 0};
  for (int e = t; e < EE; e += 256) {
#pragma unroll
    for (int c = 0; c < 4; ++c) {
      float v = einp[e * 4 + c];
      s[c] += v;
      q[c] += v * v;
    }
  }
#pragma unroll
  for (int c = 0; c < 4; ++c) {
    red[c][t] = s[c];
    red[4 + c][t] = q[c];
  }
  __syncthreads();
  for (int st = 128; st > 0; st >>= 1) {
    if (t < st) {
#pragma unroll
      for (int i = 0; i < 8; ++i) red[i][t] += red[i][t + st];
    }
    __syncthreads();
  }
  if (t < 4) {
    float mu = red[t][0] / (float)EE;
    float var = (red[4 + t][0] - (float)EE * mu * mu) / (float)(EE - 1);
    W[OFF_MU + t] = mu;
    W[OFF_RSD + t] = rsqrtf(var);
  }
}

// ------------------------------- 2) edge MLP layer-1 batchnorm statistics
// thread = (half, feature); register accumulation over a block's edge range.
__global__ void k_emlp_stats(const float* __restrict__ einp,
                             const float* __restrict__ w1,
                             const float* __restrict__ b1,
                             float* __restrict__ W) {
  int t = threadIdx.x;
  int f = t & 127;
  int half = t >> 7;
  float wa = w1[f], wb = w1[128 + f], bb = b1[f];
  float mu0 = W[OFF_MU + half * 2], mu1 = W[OFF_MU + half * 2 + 1];
  float r0 = W[OFF_RSD + half * 2], r1 = W[OFF_RSD + half * 2 + 1];
  int e0 = blockIdx.x * 512;
  float s = 0.f, q = 0.f;
  for (int e = e0; e < e0 + 512; ++e) {
    float x0 = einp[e * 4 + half * 2 + 0];
    float x1 = einp[e * 4 + half * 2 + 1];
    float z0 = (x0 - mu0) * r0;
    float z1 = (x1 - mu1) * r1;
    float h = fmaxf(z0 * wa + z1 * wb + bb, 0.f);
    s += h;
    q += h * h;
  }
  atomicAdd(&W[OFF_ESUM + half * 128 + f], s);
  atomicAdd(&W[OFF_ESQ + half * 128 + f], q);
}

// fold BN into scale/shift (biased var, eps=1e-5), per half.
__global__ void k_emlp_finalize(const float* __restrict__ g,
                                const float* __restrict__ b,
                                float* __restrict__ W) {
  int t = threadIdx.x;  // 256 = 2 halves x 128 feats
  int f = t & 127;
  float mean = W[OFF_ESUM + t] / (float)EE;
  float var = W[OFF_ESQ + t] / (float)EE - mean * mean;
  float rstd = rsqrtf(var + BN_EPS);
  float sc = rstd * g[f];
  W[OFF_ESCALE + t] = sc;
  W[OFF_ESHIFT + t] = b[f] - mean * sc;
}

// --------- 3) edge MLP layer-2 GEMM (WMMA fp32) + cosine -> edge weights
// block = 16 edges, 256 threads = 8 waves; wave w owns output cols [16w,16w+16)
__global__ void k_edge_gemm(const float* __restrict__ einp,
                            const float* __restrict__ w1,
                            const float* __restrict__ b1,
                            const float* __restrict__ w2,
                            const float* __restrict__ b2,
                            const float* __restrict__ W,
                            float* __restrict__ ew) {
  __shared__ float sh[2][16][HE];  // BN'd layer-1 activations for 16 edges
  __shared__ float sdot[16], sn1[16], sn2[16];
  int t = threadIdx.x;
  if (t < 16) { sdot[t] = 0.f; sn1[t] = 0.f; sn2[t] = 0.f; }

  // stage hbn: 2*16*128 = 4096 entries, 16 per thread (recompute layer-1)
  for (int j = 0; j < 16; ++j) {
    int lin = t + 256 * j;
    int half = lin >> 11;
    int rem = lin & 2047;
    int el = rem >> 7;
    int f = rem & 127;
    int e = blockIdx.x * 16 + el;
    float x0 = einp[e * 4 + half * 2 + 0];
    float x1 = einp[e * 4 + half * 2 + 1];
    float z0 = (x0 - W[OFF_MU + half * 2]) * W[OFF_RSD + half * 2];
    float z1 = (x1 - W[OFF_MU + half * 2 + 1]) * W[OFF_RSD + half * 2 + 1];
    float h = fmaxf(z0 * w1[f] + z1 * w1[128 + f] + b1[f], 0.f);
    sh[half][el][f] = h * W[OFF_ESCALE + half * 128 + f] +
                      W[OFF_ESHIFT + half * 128 + f];
  }
  __syncthreads();

  int wave = t >> 5, lane = t & 31;
  int n0 = wave * 16;
  int row = lane & 15;
  int koff = (lane >= 16) ? 2 : 0;
  float bias = b2[n0 + row];

  v8f acc1 = {};
  v8f acc2 = {};
#pragma unroll 4
  for (int kk = 0; kk < 32; ++kk) {
    int k = kk * 4 + koff;
    v2f a1 = {sh[0][row][k], sh[0][row][k + 1]};
    v2f a2 = {sh[1][row][k], sh[1][row][k + 1]};
    v2f bf = {w2[k * HE + n0 + row], w2[(k + 1) * HE + n0 + row]};
    acc1 = wmma4(a1, bf, acc1);
    acc2 = wmma4(a2, bf, acc2);
  }
  int mbase = (lane >= 16) ? 8 : 0;
#pragma unroll
  for (int r = 0; r < 8; ++r) {
    int m = r + mbase;
    float o1 = acc1[r] + bias;
    float o2 = acc2[r] + bias;
    atomicAdd(&sdot[m], o1 * o2);
    atomicAdd(&sn1[m], o1 * o1);
    atomicAdd(&sn2[m], o2 * o2);
  }
  __syncthreads();
  if (t < 16) {
    float denom = fmaxf(sqrtf(sn1[t]) * sqrtf(sn2[t]), COS_EPS);
    float cosv = sdot[t] / denom;
    ew[blockIdx.x * 16 + t] = (cosv + 1.f) * 0.5f;
  }
}

// ------------------------------------------------ 4) gcn_norm degree / dinv
__global__ void k_deg_init(float* deg) {
  int t = blockIdx.x * blockDim.x + threadIdx.x;
  if (t < NN) deg[t] = 1.0f;  // self-loop weight
}
__global__ void k_deg_acc(const int* __restrict__ eidx,
                          const float* __restrict__ ew, float* deg) {
  int e = blockIdx.x * blockDim.x + threadIdx.x;
  if (e < EE) atomicAdd(&deg[eidx[EE + e]], ew[e]);
}
__global__ void k_deg_fin(float* deg) {
  int t = blockIdx.x * blockDim.x + threadIdx.x;
  if (t < NN) {
    float d = deg[t];
    deg[t] = (d > 0.f) ? rsqrtf(d) : 0.f;  // now holds dinv
  }
}

// --------------- 5) layer-0 GEMM: xw = embeddings @ w0, WMMA fp32, K-split
// grid (375, 16); one wave/block. A (16xKC) and B (KCx16) tiles staged into
// LDS with async global->LDS b128 copies (coalesced, ASYNCcnt-tracked), then
// consumed by fp32 WMMA. 477 MB of A streams exactly once -> BW-bound.
#define KC 64
#define NSTAGE 311            // ceil(19900/64); last stage carries 60 cols
#define SPB 20                // stages per block (16 K-split blocks)
__global__ void k_gemm0(const float* __restrict__ emb,
                        const float* __restrict__ w0, float* __restrict__ xw) {
  __shared__ float As[16][KC];
  __shared__ float Bs[KC * 16];
  int lane = threadIdx.x;
  int m0 = blockIdx.x * 16;
  int row = lane & 15;
  int hi = lane >> 4;                 // 0/1 : lane half
  int koff = hi ? 2 : 0;
  int s0 = blockIdx.y * SPB;
  int s1 = min(NSTAGE, s0 + SPB);

  v8f acc = {};
  for (int s = s0; s < s1; ++s) {
    int k0 = s * KC;
    int kc = min(KC, INF - k0);       // 64, or 60 on the final stage
    // A tile: 16 rows x kc cols; per async op: half-wave covers 256B of a row
#pragma unroll
    for (int j = 0; j < 8; ++j) {
      int r = 2 * j + hi;
      int c4 = (lane & 15) * 4;
      if (c4 < kc)
        cp16_async(emb + (size_t)(m0 + r) * INF + k0 + c4, &As[r][c4]);
    }
    // B tile: w0 rows k0..k0+kc (16 floats each) -> contiguous kc*64 bytes
#pragma unroll
    for (int j = 0; j < 8; ++j) {
      int c4 = (j * 32 + lane) * 4;
      if (c4 < kc * 16) cp16_async(w0 + (size_t)k0 * HH + c4, &Bs[c4]);
    }
    async_wait0();
    int nks = kc >> 2;                // 16 or 15 WMMA k-steps
    for (int ks = 0; ks < nks; ++ks) {
      int k = ks * 4 + koff;
      v2f a = {As[row][k], As[row][k + 1]};
      v2f b = {Bs[k * HH + row], Bs[(k + 1) * HH + row]};
      acc = wmma4(a, b, acc);
    }
  }
  int mbase = hi ? 8 : 0;
#pragma unroll
  for (int r = 0; r < 8; ++r)
    atomicAdd(&xw[(m0 + r + mbase) * HH + row], acc[r]);
}

// --------------------- 6) hidden-layer GEMM: xw = x(6000x16) @ w(16x16)
__global__ void k_gemm_h(const float* __restrict__ x,
                         const float* __restrict__ w, float* __restrict__ xw) {
  int lane = threadIdx.x;
  int m0 = blockIdx.x * 16;
  int row = lane & 15;
  int koff = (lane >= 16) ? 2 : 0;
  v8f acc = {};
#pragma unroll
  for (int kk = 0; kk < 4; ++kk) {
    int k = kk * 4 + koff;
    v2f a = {x[(m0 + row) * HH + k], x[(m0 + row) * HH + k + 1]};
    v2f b = {w[k * HH + row], w[(k + 1) * HH + row]};
    acc = wmma4(a, b, acc);
  }
  int mbase = (lane >= 16) ? 8 : 0;
#pragma unroll
  for (int r = 0; r < 8; ++r)
    xw[(m0 + r + mbase) * HH + row] = acc[r];
}

// ------------------------------------- 7) graph aggregation (scatter-add)
__global__ void k_agg_init(const float* __restrict__ xw,
                           const float* __restrict__ dinv,
                           const float* __restrict__ bias,
                           float* __restrict__ agg) {
  int t = blockIdx.x * blockDim.x + threadIdx.x;  // N*H
  int n = t >> 4, h = t & 15;
  float di = dinv[n];
  agg[t] = di * di * xw[t] + bias[h];  // self-loop + gcn bias
}
__global__ void k_agg_edge(const int* __restrict__ eidx,
                           const float* __restrict__ ew,
                           const float* __restrict__ dinv,
                           const float* __restrict__ xw,
                           float* __restrict__ agg) {
  int t = blockIdx.x * blockDim.x + threadIdx.x;  // E*H
  int e = t >> 4, h = t & 15;
  int r = eidx[e], c = eidx[EE + e];
  float nw = dinv[r] * ew[e] * dinv[c];
  atomicAdd(&agg[c * HH + h], nw * xw[r * HH + h]);
}

// ----------------------------------- 8) GCN batchnorm (single block, 16 f)
__global__ void k_bn_stats(const float* __restrict__ agg,
                           const float* __restrict__ g,
                           const float* __restrict__ b,
                           float* __restrict__ W) {
  __shared__ float ssum[256], ssq[256];
  int t = threadIdx.x;
  int f = t & 15, sl = t >> 4;
  float s = 0.f, q = 0.f;
  for (int n = sl; n < NN; n += 16) {
    float v = agg[n * HH + f];
    s += v;
    q += v * v;
  }
  ssum[t] = s;
  ssq[t] = q;
  __syncthreads();
  for (int st = 128; st >= 16; st >>= 1) {
    if (t < st) { ssum[t] += ssum[t + st]; ssq[t] += ssq[t + st]; }
    __syncthreads();
  }
  if (t < 16) {
    float mean = ssum[t] / (float)NN;
    float var = ssq[t] / (float)NN - mean * mean;
    float rstd = rsqrtf(var + BN_EPS);
    float sc = rstd * g[t];
    W[OFF_BNSCALE + t] = sc;
    W[OFF_BNSHIFT + t] = b[t] - mean * sc;
  }
}
__global__ void k_bn_apply(const float* __restrict__ agg,
                           const float* __restrict__ W,
                           const float* __restrict__ prev, int has_prev,
                           float* __restrict__ out) {
  int t = blockIdx.x * blockDim.x + threadIdx.x;  // N*H
  int h = t & 15;
  float v = fmaxf(agg[t] * W[OFF_BNSCALE + h] + W[OFF_BNSHIFT + h], 0.f);
  if (has_prev) v += 0.7f * prev[t];
  out[t] = v;
}

// ------------------------------------------------ 9) softmax + projection
__global__ void k_softmax(const float* __restrict__ lw, float* __restrict__ W) {
  if (threadIdx.x == 0) {
    float m = fmaxf(fmaxf(lw[0], lw[1]), fmaxf(lw[2], lw[3]));
    float e0 = expf(lw[0] - m), e1 = expf(lw[1] - m);
    float e2 = expf(lw[2] - m), e3 = expf(lw[3] - m);
    float s = e0 + e1 + e2 + e3;
    W[OFF_WSM + 0] = e0 / s;
    W[OFF_WSM + 1] = e1 / s;
    W[OFF_WSM + 2] = e2 / s;
    W[OFF_WSM + 3] = e3 / s;
  }
}
__global__ void k_out(const float* __restrict__ lo,  // [4][N*H]
                      const float* __restrict__ W,
                      const float* __restrict__ pw,
                      const float* __restrict__ pb, float* __restrict__ out) {
  int n = blockIdx.x * blockDim.x + threadIdx.x;
  if (n >= NN) return;
  float w0 = W[OFF_WSM + 0], w1 = W[OFF_WSM + 1];
  float w2 = W[OFF_WSM + 2], w3 = W[OFF_WSM + 3];
  float c0 = pb[0], c1 = pb[1];
#pragma unroll
  for (int h = 0; h < HH; ++h) {
    int idx = n * HH + h;
    float e = w0 * lo[idx] + w1 * lo[NN * HH + idx] +
              w2 * lo[2 * NN * HH + idx] + w3 * lo[3 * NN * HH + idx];
    c0 += e * pw[h * CC + 0];
    c1 += e * pw[h * CC + 1];
  }
  out[n * CC + 0] = c0;
  out[n * CC + 1] = c1;
}

// ---------------------------------------------------------------- launcher
extern "C" void kernel_launch(void* const* d_in, const int* in_sizes, int n_in,
                              void* d_out, int out_size, void* d_ws,
                              size_t ws_size, hipStream_t stream) {
  const float* emb   = (const float*)d_in[0];
  const int*   eidx  = (const int*)d_in[1];
  const float* einp  = (const float*)d_in[2];
  const float* e_w1  = (const float*)d_in[3];
  const float* e_b1  = (const float*)d_in[4];
  const float* e_bng = (const float*)d_in[5];
  const float* e_bnb = (const float*)d_in[6];
  const float* e_w2  = (const float*)d_in[7];
  const float* e_b2  = (const float*)d_in[8];
  const float* gw0   = (const float*)d_in[9];
  const float* gwr   = (const float*)d_in[10];
  const float* gcnb  = (const float*)d_in[11];
  const float* bng   = (const float*)d_in[12];
  const float* bnb   = (const float*)d_in[13];
  const float* lw    = (const float*)d_in[14];
  const float* pw    = (const float*)d_in[15];
  const float* pb    = (const float*)d_in[16];
  float* out = (float*)d_out;
  float* W = (float*)d_ws;

  float* ew   = W + OFF_BIG;            // [E]
  float* dinv = ew + EE;                // [N] (deg -> dinv in place)
  float* xw   = dinv + NN + 16;         // [N*H]
  float* agg  = xw + NN * HH;           // [N*H]
  float* lo   = agg + NN * HH;          // [4][N*H]

  // zero stats + xw (atomic accumulation targets)
  k_zero<<<8, 256, 0, stream>>>(W, 2048);
  k_zero<<<(NN * HH) / 256, 256, 0, stream>>>(xw, NN * HH);

  // edge-input standardization + edge MLP stats
  k_edge_std<<<1, 256, 0, stream>>>(einp, W);
  k_emlp_stats<<<EE / 512, 256, 0, stream>>>(einp, e_w1, e_b1, W);
  k_emlp_finalize<<<1, 256, 0, stream>>>(e_bng, e_bnb, W);

  // edge MLP layer-2 GEMM (WMMA) + cosine -> edge weights
  k_edge_gemm<<<EE / 16, 256, 0, stream>>>(einp, e_w1, e_b1, e_w2, e_b2, W, ew);

  // gcn_norm degrees
  k_deg_init<<<(NN + 255) / 256, 256, 0, stream>>>(dinv);
  k_deg_acc<<<EE / 256, 256, 0, stream>>>(eidx, ew, dinv);
  k_deg_fin<<<(NN + 255) / 256, 256, 0, stream>>>(dinv);

  // layer 0 big GEMM (BW-bound; async LDS staging + K-split WMMA)
  k_gemm0<<<dim3(NN / 16, 16), 32, 0, stream>>>(emb, gw0, xw);

  for (int i = 0; i < LL; ++i) {
    if (i > 0)
      k_gemm_h<<<NN / 16, 32, 0, stream>>>(lo + (i - 1) * NN * HH,
                                           gwr + (i - 1) * HH * HH, xw);
    k_agg_init<<<(NN * HH) / 256, 256, 0, stream>>>(xw, dinv, gcnb + i * HH,
                                                    agg);
    k_agg_edge<<<(EE * HH) / 256, 256, 0, stream>>>(eidx, ew, dinv, xw, agg);
    k_bn_stats<<<1, 256, 0, stream>>>(agg, bng + i * HH, bnb + i * HH, W);
    k_bn_apply<<<(NN * HH) / 256, 256, 0, stream>>>(
        agg, W, (i > 0) ? (lo + (i - 1) * NN * HH) : nullptr, i > 0,
        lo + i * NN * HH);
  }

  k_softmax<<<1, 32, 0, stream>>>(lw, W);
  k_out<<<(NN + 255) / 256, 256, 0, stream>>>(lo, W, pw, pb, out);
}